// BiLSTM_CRF_5531917877819
// MI455X (gfx1250) — compile-verified
//
#include <hip/hip_runtime.h>
#include <hip/hip_bf16.h>

typedef float v2f __attribute__((ext_vector_type(2)));
typedef float v8f __attribute__((ext_vector_type(8)));

namespace {
constexpr int L_TAGS   = 12;
constexpr int START_ID = 10;
constexpr int END_ID   = 11;
constexpr int T_SEQ    = 512;
constexpr int B_BATCH  = 64;
constexpr int E_DIM    = 512;
constexpr int H_DIM    = 512;
constexpr int G4       = 2048;              // 4*H
constexpr int M_ROWS   = B_BATCH * T_SEQ;   // 32768
constexpr int LDA      = 36;                // LDS row stride (floats):
                                            // 144B = 16B aligned rows, and
                                            // 36*r mod 64 hits 16 distinct
                                            // banks -> conflict-free frags
constexpr int A_SZ     = 128 * LDA;         // one A stage
constexpr int B_SZ     = 64 * LDA;          // one B stage
}

// Async global->LDS path (CDNA5 GLOBAL_LOAD_ASYNC_TO_LDS_B128, ASYNCcnt).
#if defined(__HIP_DEVICE_COMPILE__) && \
    __has_builtin(__builtin_amdgcn_global_load_async_to_lds_b128) && \
    __has_builtin(__builtin_amdgcn_s_wait_asynccnt)
#define USE_ASYNC_LDS 1
#else
#define USE_ASYNC_LDS 0
#endif

#if USE_ASYNC_LDS
// Builtin signature (from gfx1250 device-pass diagnostics):
//   void (int __vector(4) addrspace(1)*, int __vector(4) addrspace(3)*,
//         imm int offset, imm int cpol)
typedef int v4i_t __attribute__((vector_size(16)));
typedef __attribute__((address_space(1))) v4i_t gv4i;
typedef __attribute__((address_space(3))) v4i_t lv4i;
__device__ __forceinline__ void async_copy16(const float* g, float* l) {
  // 16 bytes per lane, global -> LDS, default scope/temporal policy
  __builtin_amdgcn_global_load_async_to_lds_b128((gv4i*)g, (lv4i*)l, 0, 0);
}
#endif

__device__ __forceinline__ void wait_async_all() {
#if USE_ASYNC_LDS
  __builtin_amdgcn_s_wait_asynccnt(0);
#endif
}

// ---- fp32 WMMA helpers (CDNA5 V_WMMA_F32_16X16X4_F32) -----------------------
// A: 16x4 (M x K).  lanes 0-15: row=lane,   VGPR0=K0, VGPR1=K1
//                   lanes16-31: row=lane-16,VGPR0=K2, VGPR1=K3
// B: 4x16 (K x N), mirrored with N on lane&15.
// C/D 16x16: VGPR e: lanes0-15 -> M=e, lanes16-31 -> M=8+e; N = lane&15.
// Builtin restricted to the device pass (host aux-builtin handling is
// unreliable on this toolchain; host never codegens __device__ bodies).
__device__ __forceinline__ v8f wmma_f32(v2f a, v2f b, v8f c) {
#if defined(__HIP_DEVICE_COMPILE__)
  return __builtin_amdgcn_wmma_f32_16x16x4_f32(false, a, false, b,
                                               (short)0, c, false, false);
#else
  return c;
#endif
}

__device__ __forceinline__ v2f ldfrag(const float* p, int ld, int lane) {
  const int r  = lane & 15;
  const int kk = (lane >> 4) << 1;
  const float* q = p + (size_t)r * (size_t)ld + kk;
  v2f f; f.x = q[0]; f.y = q[1]; return f;
}

__device__ __forceinline__ float sigm(float x) {
  return 1.0f / (1.0f + __expf(-x));
}

// ---- small utility ----------------------------------------------------------
__global__ void k_zero(float* __restrict__ p, int n) {
  int i = blockIdx.x * blockDim.x + threadIdx.x;
  if (i < n) p[i] = 0.0f;
}

// ---- tile stagers for the input GEMM ---------------------------------------
// A tile: 128 rows x 32 cols, gathered through sents.
// B tile: 64 rows (w_ih rows n0..n0+63) x 32 cols.
__device__ __forceinline__ void stage_tiles(
    const int* __restrict__ sents, const float* __restrict__ emb,
    const float* __restrict__ w_ih, int m0, int n0, int K0, int tid,
    float* __restrict__ As, float* __restrict__ Bs) {
#if USE_ASYNC_LDS
  {
    const int r  = tid >> 1;           // 0..127
    const int c0 = (tid & 1) * 16;     // 0 or 16
    const int tok = sents[m0 + r];
    const float* g = emb + (size_t)tok * E_DIM + K0 + c0;
    float* l = &As[r * LDA + c0];
    async_copy16(g + 0,  l + 0);
    async_copy16(g + 4,  l + 4);
    async_copy16(g + 8,  l + 8);
    async_copy16(g + 12, l + 12);
  }
  {
    const int r  = tid >> 2;           // 0..63
    const int c0 = (tid & 3) * 8;      // 0,8,16,24
    const float* g = w_ih + (size_t)(n0 + r) * E_DIM + K0 + c0;
    float* l = &Bs[r * LDA + c0];
    async_copy16(g + 0, l + 0);
    async_copy16(g + 4, l + 4);
  }
#else
  {
    const int r  = tid >> 1;
    const int c0 = (tid & 1) * 16;
    const int tok = sents[m0 + r];
    const float4* src = reinterpret_cast<const float4*>(
        emb + (size_t)tok * E_DIM + K0 + c0);
    float* dst = &As[r * LDA + c0];
    float4 v0 = src[0], v1 = src[1], v2 = src[2], v3 = src[3];
    dst[0]  = v0.x; dst[1]  = v0.y; dst[2]  = v0.z; dst[3]  = v0.w;
    dst[4]  = v1.x; dst[5]  = v1.y; dst[6]  = v1.z; dst[7]  = v1.w;
    dst[8]  = v2.x; dst[9]  = v2.y; dst[10] = v2.z; dst[11] = v2.w;
    dst[12] = v3.x; dst[13] = v3.y; dst[14] = v3.z; dst[15] = v3.w;
  }
  {
    const int r  = tid >> 2;
    const int c0 = (tid & 3) * 8;
    const float4* src = reinterpret_cast<const float4*>(
        w_ih + (size_t)(n0 + r) * E_DIM + K0 + c0);
    float* dst = &Bs[r * LDA + c0];
    float4 v0 = src[0], v1 = src[1];
    dst[0] = v0.x; dst[1] = v0.y; dst[2] = v0.z; dst[3] = v0.w;
    dst[4] = v1.x; dst[5] = v1.y; dst[6] = v1.z; dst[7] = v1.w;
  }
#endif
}

// ---- fused embedding-gather + input projection GEMM -------------------------
// xg[m][n] = sum_k emb[sents[m]][k] * w_ih[n][k] + bias[n]
// M=32768, N=2048, K=512.  Block tile 128x64, Ktile=32, 8 waves (4x2),
// each wave computes a 32x32 patch = 2x2 fp32 WMMA tiles.
// Double-buffered LDS; stage k+1 is fetched asynchronously (ASYNCcnt)
// while WMMAs consume stage k.
__global__ __launch_bounds__(256) void k_input_gemm(
    const int*   __restrict__ sents,
    const float* __restrict__ emb,
    const float* __restrict__ w_ih,
    const float* __restrict__ bias,
    float*       __restrict__ xg) {
  __shared__ float As[2 * A_SZ];
  __shared__ float Bs[2 * B_SZ];

  const int m0   = blockIdx.x * 128;
  const int n0   = blockIdx.y * 64;
  const int tid  = threadIdx.x;
  const int lane = tid & 31;
  const int wid  = tid >> 5;
  const int wm   = wid & 3;    // 0..3  (M direction)
  const int wn   = wid >> 2;   // 0..1  (N direction)

  const v8f vzero = {0.f, 0.f, 0.f, 0.f, 0.f, 0.f, 0.f, 0.f};
  v8f acc[2][2];
  acc[0][0] = vzero; acc[0][1] = vzero; acc[1][0] = vzero; acc[1][1] = vzero;

  // prologue: stage 0
  stage_tiles(sents, emb, w_ih, m0, n0, 0, tid, As, Bs);
  wait_async_all();
  __syncthreads();

  int cur = 0;
  for (int K0 = 0; K0 < E_DIM; K0 += 32) {
    // prefetch next stage into the other buffer
    if (K0 + 32 < E_DIM) {
      stage_tiles(sents, emb, w_ih, m0, n0, K0 + 32, tid,
                  As + (cur ^ 1) * A_SZ, Bs + (cur ^ 1) * B_SZ);
    }

    const float* Ac = As + cur * A_SZ;
    const float* Bc = Bs + cur * B_SZ;
#pragma unroll
    for (int kk = 0; kk < 32; kk += 4) {
      v2f a0 = ldfrag(&Ac[(wm * 32 + 0)  * LDA + kk], LDA, lane);
      v2f a1 = ldfrag(&Ac[(wm * 32 + 16) * LDA + kk], LDA, lane);
      v2f b0 = ldfrag(&Bc[(wn * 32 + 0)  * LDA + kk], LDA, lane);
      v2f b1 = ldfrag(&Bc[(wn * 32 + 16) * LDA + kk], LDA, lane);
      acc[0][0] = wmma_f32(a0, b0, acc[0][0]);
      acc[0][1] = wmma_f32(a0, b1, acc[0][1]);
      acc[1][0] = wmma_f32(a1, b0, acc[1][0]);
      acc[1][1] = wmma_f32(a1, b1, acc[1][1]);
    }

    wait_async_all();   // own async ops landed in LDS
    __syncthreads();    // workgroup-wide visibility + reuse safety
    cur ^= 1;
  }

  const int nn = lane & 15;
  const int mb = (lane >> 4) * 8;
#pragma unroll
  for (int ti = 0; ti < 2; ++ti) {
#pragma unroll
    for (int tj = 0; tj < 2; ++tj) {
      const int gm = m0 + wm * 32 + ti * 16 + mb;
      const int gn = n0 + wn * 32 + tj * 16 + nn;
      const float bv = bias[gn];
#pragma unroll
      for (int e = 0; e < 8; ++e) {
        xg[(size_t)(gm + e) * G4 + gn] = acc[ti][tj][e] + bv;
      }
    }
  }
}

// ---- one LSTM time step (GEMM + gates fused per wave) -----------------------
// Each wave owns a 16(batch) x 16(hidden) tile and computes all 4 gate tiles,
// then updates c, h and writes hs[:, t, :].  128 waves total; w_hh and h stay
// L2-resident across the 512 sequential steps.
__global__ __launch_bounds__(256) void k_lstm_step(
    const float* __restrict__ xg,      // (B,T,4H)
    const float* __restrict__ w_hh,    // (4H,H) row-major
    const float* __restrict__ hprev,   // batch-major matrix, leading dim ldh
    int ldh,
    float*       __restrict__ cbuf,    // (B,H)
    float*       __restrict__ hs,      // (B,T,H)
    int t) {
  const int lane = threadIdx.x & 31;
  const int wid  = threadIdx.x >> 5;
  const int wgl  = blockIdx.x * 8 + wid;     // 0..127
  const int m0   = (wgl & 3) * 16;           // batch tile
  const int n0   = (wgl >> 2) * 16;          // hidden tile

  const v8f vzero = {0.f, 0.f, 0.f, 0.f, 0.f, 0.f, 0.f, 0.f};
  v8f ai = vzero, af = vzero, ag = vzero, ao = vzero;

  const float* hbase = hprev + (size_t)m0 * (size_t)ldh;
  const float* wi = w_hh + (size_t)(0 * H_DIM + n0) * H_DIM;
  const float* wf = w_hh + (size_t)(1 * H_DIM + n0) * H_DIM;
  const float* wg = w_hh + (size_t)(2 * H_DIM + n0) * H_DIM;
  const float* wo = w_hh + (size_t)(3 * H_DIM + n0) * H_DIM;

  for (int k = 0; k < H_DIM; k += 4) {
    v2f a  = ldfrag(hbase + k, ldh, lane);
    v2f bi = ldfrag(wi + k, H_DIM, lane);
    v2f bf = ldfrag(wf + k, H_DIM, lane);
    v2f bg = ldfrag(wg + k, H_DIM, lane);
    v2f bo = ldfrag(wo + k, H_DIM, lane);
    ai = wmma_f32(a, bi, ai);
    af = wmma_f32(a, bf, af);
    ag = wmma_f32(a, bg, ag);
    ao = wmma_f32(a, bo, ao);
  }

  const int nn = lane & 15;
  const int mb = (lane >> 4) * 8;
#pragma unroll
  for (int e = 0; e < 8; ++e) {
    const int bb = m0 + mb + e;
    const int j  = n0 + nn;
    const size_t xbase = ((size_t)bb * T_SEQ + t) * G4;
    const float gi = ai[e] + xg[xbase + 0 * H_DIM + j];
    const float gf = af[e] + xg[xbase + 1 * H_DIM + j];
    const float gg = ag[e] + xg[xbase + 2 * H_DIM + j];
    const float go = ao[e] + xg[xbase + 3 * H_DIM + j];
    const size_t cj = (size_t)bb * H_DIM + j;
    const float c_old = cbuf[cj];
    const float c_new = sigm(gf) * c_old + sigm(gi) * tanhf(gg);
    const float h_new = sigm(go) * tanhf(c_new);
    cbuf[cj] = c_new;
    hs[((size_t)bb * T_SEQ + t) * H_DIM + j] = h_new;
  }
}

// ---- FC: feats[m][l] = sum_k tanh(hcat[m][k]) * fc_w[l][k] + fc_b[l] --------
// N padded to 16 (L=12).  One wave per 16-row tile.
__global__ __launch_bounds__(256) void k_fc(
    const float* __restrict__ hs_f,
    const float* __restrict__ hs_r,
    const float* __restrict__ fc_w,    // (12,1024)
    const float* __restrict__ fc_b,
    float*       __restrict__ feats) { // (B,T,12)
  const int lane = threadIdx.x & 31;
  const int wid  = threadIdx.x >> 5;
  const int m0   = (blockIdx.x * 8 + wid) * 16;

  const v8f vzero = {0.f, 0.f, 0.f, 0.f, 0.f, 0.f, 0.f, 0.f};
  v8f acc = vzero;

  const int r   = lane & 15;
  const int kk2 = (lane >> 4) * 2;
  const int nn  = lane & 15;

  for (int k = 0; k < 2 * H_DIM; k += 4) {
    const int ka = k + kk2;                       // pair {ka, ka+1} same half
    const float* hsrc = (ka < H_DIM) ? hs_f : hs_r;
    const int kh = ka & (H_DIM - 1);
    v2f a, b;
    a.x = tanhf(hsrc[(size_t)(m0 + r) * H_DIM + kh]);
    a.y = tanhf(hsrc[(size_t)(m0 + r) * H_DIM + kh + 1]);
    if (nn < L_TAGS) {
      b.x = fc_w[(size_t)nn * (2 * H_DIM) + ka];
      b.y = fc_w[(size_t)nn * (2 * H_DIM) + ka + 1];
    } else {
      b.x = 0.f; b.y = 0.f;
    }
    acc = wmma_f32(a, b, acc);
  }

  const int mb = (lane >> 4) * 8;
  if (nn < L_TAGS) {
    const float bb = fc_b[nn];
#pragma unroll
    for (int e = 0; e < 8; ++e) {
      feats[(size_t)(m0 + mb + e) * L_TAGS + nn] = acc[e] + bb;
    }
  }
}

// ---- CRF forward (log Z) + gold score, one wave per batch -------------------
__global__ __launch_bounds__(32) void k_crf(
    const float* __restrict__ feats,
    const float* __restrict__ trans,    // (12,12)
    const int*   __restrict__ labels,   // (B,T)
    const int*   __restrict__ lengths,  // (B,)
    float*       __restrict__ diff) {   // (B,)
  __shared__ float tr[L_TAGS * L_TAGS];
  __shared__ float alpha_s[L_TAGS];

  const int b    = blockIdx.x;
  const int lane = threadIdx.x;

  for (int i = lane; i < L_TAGS * L_TAGS; i += 32) tr[i] = trans[i];
  if (lane < L_TAGS) alpha_s[lane] = 0.0f;
  __syncthreads();

  int len = lengths[b];
  if (len > T_SEQ) len = T_SEQ;

  for (int t = 0; t < len; ++t) {
    float newv = 0.0f;
    if (lane < L_TAGS) {
      const float f = feats[((size_t)b * T_SEQ + t) * L_TAGS + lane];
      float m = -1e30f;
#pragma unroll
      for (int i = 0; i < L_TAGS; ++i)
        m = fmaxf(m, alpha_s[i] + tr[i * L_TAGS + lane]);
      float s = 0.0f;
#pragma unroll
      for (int i = 0; i < L_TAGS; ++i)
        s += __expf(alpha_s[i] + tr[i * L_TAGS + lane] - m);
      newv = m + __logf(s) + f;
    }
    __syncthreads();
    if (lane < L_TAGS) alpha_s[lane] = newv;
    __syncthreads();
  }

  // log Z = LSE_j(alpha[j] + trans[j][END])
  float v = (lane < L_TAGS) ? (alpha_s[lane] + tr[lane * L_TAGS + END_ID])
                            : -1e30f;
  float m = v;
  for (int off = 16; off; off >>= 1) m = fmaxf(m, __shfl_xor(m, off));
  float e = (lane < L_TAGS) ? __expf(v - m) : 0.0f;
  float s = e;
  for (int off = 16; off; off >>= 1) s += __shfl_xor(s, off);
  const float logz = m + __logf(s);

  // gold score, t-parallel over lanes
  float part = 0.0f;
  for (int t = lane; t < len; t += 32) {
    const int nxt  = labels[(size_t)b * T_SEQ + t];
    const int prev = (t == 0) ? START_ID : labels[(size_t)b * T_SEQ + t - 1];
    part += feats[((size_t)b * T_SEQ + t) * L_TAGS + nxt]
          + tr[prev * L_TAGS + nxt];
  }
  for (int off = 16; off; off >>= 1) part += __shfl_xor(part, off);

  if (lane == 0) {
    const int last = labels[(size_t)b * T_SEQ + len - 1];
    const float gold = part + tr[last * L_TAGS + END_ID];
    diff[b] = logz - gold;
  }
}

__global__ __launch_bounds__(64) void k_reduce(const float* __restrict__ diff,
                                               float* __restrict__ out) {
  __shared__ float sh[64];
  const int lane = threadIdx.x;
  sh[lane] = diff[lane];
  __syncthreads();
  if (lane < 32) {
    float s = sh[lane] + sh[lane + 32];
    for (int off = 16; off; off >>= 1) s += __shfl_xor(s, off);
    if (lane == 0) out[0] = s;
  }
}

// ---- host-side orchestration ------------------------------------------------
extern "C" void kernel_launch(void* const* d_in, const int* in_sizes, int n_in,
                              void* d_out, int out_size, void* d_ws,
                              size_t ws_size, hipStream_t stream) {
  (void)in_sizes; (void)n_in; (void)out_size; (void)ws_size;

  const int*   sents   = (const int*)  d_in[0];
  const int*   labels  = (const int*)  d_in[1];
  const int*   lengths = (const int*)  d_in[2];
  const float* emb     = (const float*)d_in[3];
  const float* w_ih_f  = (const float*)d_in[4];
  const float* w_hh_f  = (const float*)d_in[5];
  const float* b_f     = (const float*)d_in[6];
  const float* w_ih_r  = (const float*)d_in[7];
  const float* w_hh_r  = (const float*)d_in[8];
  const float* b_r     = (const float*)d_in[9];
  const float* fc_w    = (const float*)d_in[10];
  const float* fc_b    = (const float*)d_in[11];
  const float* trans   = (const float*)d_in[12];

  float* ws = (float*)d_ws;
  size_t off = 0;
  float* xg    = ws + off; off += (size_t)M_ROWS * G4;      // 256 MB (reused)
  float* hs_f  = ws + off; off += (size_t)M_ROWS * H_DIM;   //  64 MB
  float* hs_r  = ws + off; off += (size_t)M_ROWS * H_DIM;   //  64 MB
  float* cbuf  = ws + off; off += (size_t)B_BATCH * H_DIM;
  float* zbuf  = ws + off; off += (size_t)B_BATCH * H_DIM;
  float* feats = ws + off; off += (size_t)M_ROWS * L_TAGS;
  float* diff  = ws + off; off += B_BATCH;

  const int hc = B_BATCH * H_DIM;
  k_zero<<<dim3((hc + 255) / 256), 256, 0, stream>>>(zbuf, hc);

  // ---------- forward direction ----------
  k_input_gemm<<<dim3(M_ROWS / 128, G4 / 64), 256, 0, stream>>>(
      sents, emb, w_ih_f, b_f, xg);
  k_zero<<<dim3((hc + 255) / 256), 256, 0, stream>>>(cbuf, hc);
  for (int t = 0; t < T_SEQ; ++t) {
    const float* hprev = (t == 0) ? zbuf : (hs_f + (size_t)(t - 1) * H_DIM);
    const int    ldh   = (t == 0) ? H_DIM : T_SEQ * H_DIM;
    k_lstm_step<<<dim3(16), 256, 0, stream>>>(xg, w_hh_f, hprev, ldh,
                                              cbuf, hs_f, t);
  }

  // ---------- reverse direction (xg buffer reused) ----------
  k_input_gemm<<<dim3(M_ROWS / 128, G4 / 64), 256, 0, stream>>>(
      sents, emb, w_ih_r, b_r, xg);
  k_zero<<<dim3((hc + 255) / 256), 256, 0, stream>>>(cbuf, hc);
  for (int s = 0; s < T_SEQ; ++s) {
    const int t = T_SEQ - 1 - s;
    const float* hprev = (s == 0) ? zbuf : (hs_r + (size_t)(t + 1) * H_DIM);
    const int    ldh   = (s == 0) ? H_DIM : T_SEQ * H_DIM;
    k_lstm_step<<<dim3(16), 256, 0, stream>>>(xg, w_hh_r, hprev, ldh,
                                              cbuf, hs_r, t);
  }

  // ---------- FC + CRF + loss ----------
  k_fc<<<dim3(M_ROWS / 16 / 8), 256, 0, stream>>>(hs_f, hs_r, fc_w, fc_b,
                                                  feats);
  k_crf<<<dim3(B_BATCH), 32, 0, stream>>>(feats, trans, labels, lengths, diff);
  k_reduce<<<dim3(1), 64, 0, stream>>>(diff, (float*)d_out);
}